// GridNNGraph_69337952027017
// MI455X (gfx1250) — compile-verified
//
#include <hip/hip_runtime.h>
#include <hip/hip_bf16.h>
#include <stdint.h>

// ---------------- problem constants (from reference) ----------------
#define BB 8            // batches
#define NP 100000       // points per batch
#define CF 64           // feature channels
#define NSAMP 4096      // centers
#define NNEI 32         // neighbors per center
#define NT ((NP + 255) / 256)   // radix tiles per batch = 391

typedef unsigned int v4u __attribute__((ext_vector_type(4)));
typedef int          v8i __attribute__((ext_vector_type(8)));
typedef int          v4i __attribute__((ext_vector_type(4)));

#if defined(__gfx1250__) && __has_builtin(__builtin_amdgcn_tensor_load_to_lds)
#define USE_TDM 1
#else
#define USE_TDM 0
#endif

// =====================================================================
// k_gather FIRST in the file so the disasm snippet shows the TDM path:
// one wave32 per (b, s): rel + TDM gather-mode row fetch of feat into
// LDS (4 descriptors x 8 x 32-bit row indices), s_wait_tensorcnt, then
// stream the 32x64 feature block into the 67-stride output.
// =====================================================================
__global__ __launch_bounds__(256) void k_gather(
        const float* __restrict__ pos, const float* __restrict__ feat,
        const uint32_t* __restrict__ ctr, const uint32_t* __restrict__ srcIdx,
        float* __restrict__ out) {
    __shared__ float ldsF[8 * NNEI * CF];   // 64 KB: 8 waves x 32 rows x 64 f32
    int w = threadIdx.x >> 5, lane = threadIdx.x & 31;
    int gw = blockIdx.x * 8 + w;
    int b = gw / NSAMP, s = gw % NSAMP;
    uint32_t c = ctr[(size_t)b * NSAMP + s];
    size_t pbase = ((size_t)b * NSAMP + s) * NNEI;
    uint32_t srci = srcIdx[pbase + lane];
    const float* ps = pos + ((size_t)b * NP + srci) * 3;
    const float* pc = pos + ((size_t)b * NP + c) * 3;
    size_t obase = (pbase + lane) * 67;
    out[obase + 0] = ps[0] - pc[0];
    out[obase + 1] = ps[1] - pc[1];
    out[obase + 2] = ps[2] - pc[2];
    const float* fb = feat + (size_t)b * NP * CF;

#if USE_TDM
    // TDM gather-mode: 4 descriptors x 8 row indices (32-bit) of 64-f32 rows
    int vsrc = (int)srci;
    uint32_t lbase = (uint32_t)(uintptr_t)&ldsF[w * NNEI * CF];
    lbase = (uint32_t)__builtin_amdgcn_readfirstlane((int)lbase);
    uint64_t ga = (uint64_t)(uintptr_t)fb;
#pragma unroll
    for (int d = 0; d < 4; ++d) {
        v4u g0;
        g0[0] = 1u | (1u << 30) | (1u << 31);           // count=1, 32b idx, gather
        g0[1] = lbase + (uint32_t)(d * 8 * CF * 4);     // lds_addr (bytes)
        g0[2] = (uint32_t)ga;                           // global_addr[31:0]
        g0[3] = ((uint32_t)(ga >> 32) & 0x01FFFFFFu) | 0x80000000u; // addr hi | type=2
        v8i g1 = (v8i)0;
        g1[0] = (int)(2u << 16);                        // data_size = 4B
        g1[1] = (int)(64u << 16);                       // tensor_dim0 = 64 (lo16)
        g1[2] = (int)(((uint32_t)NP & 0xFFFFu) << 16);  // tensor_dim1 lo16
        g1[3] = (int)(((uint32_t)NP >> 16) | (64u << 16)); // dim1 hi16 | tile_dim0=64
        g1[4] = 8;                                      // tile_dim1 = #indices
        g1[5] = 64;                                     // tensor_dim0_stride = 64
        v4i g2, g3;
        g2[0] = __builtin_amdgcn_readlane(vsrc, d * 8 + 0);
        g2[1] = __builtin_amdgcn_readlane(vsrc, d * 8 + 1);
        g2[2] = __builtin_amdgcn_readlane(vsrc, d * 8 + 2);
        g2[3] = __builtin_amdgcn_readlane(vsrc, d * 8 + 3);
        g3[0] = __builtin_amdgcn_readlane(vsrc, d * 8 + 4);
        g3[1] = __builtin_amdgcn_readlane(vsrc, d * 8 + 5);
        g3[2] = __builtin_amdgcn_readlane(vsrc, d * 8 + 6);
        g3[3] = __builtin_amdgcn_readlane(vsrc, d * 8 + 7);
#if defined(__clang_major__) && __clang_major__ >= 23
        __builtin_amdgcn_tensor_load_to_lds(g0, g1, g2, g3, (v8i)0, 0);
#else
        __builtin_amdgcn_tensor_load_to_lds(g0, g1, g2, g3, 0);
#endif
    }
    __builtin_amdgcn_s_wait_tensorcnt(0);
    // compiler barrier: TDM wrote LDS behind the optimizer's back — keep the
    // LDS reads below from being scheduled above the tensorcnt drain.
    asm volatile("" ::: "memory");
    for (int t = lane; t < NNEI * CF; t += 32) {
        int row = t >> 6, col = t & 63;
        out[(pbase + row) * 67 + 3 + col] = ldsF[w * NNEI * CF + t];
    }
#else
    // fallback: direct gather (no TDM builtin on this toolchain)
    for (int t = lane; t < NNEI * CF; t += 32) {
        int row = t >> 6, col = t & 63;
        uint32_t rs = srcIdx[pbase + row];
        out[(pbase + row) * 67 + 3 + col] = fb[(size_t)rs * CF + col];
    }
#endif
}

// ---------------- threefry2x32 (JAX-compatible schedule) ----------------
__device__ __forceinline__ uint32_t rotl32(uint32_t x, int r) {
    return (x << r) | (x >> (32 - r));
}

__device__ __forceinline__ void threefry2x32(uint32_t k0, uint32_t k1,
                                             uint32_t x0, uint32_t x1,
                                             uint32_t& o0, uint32_t& o1) {
    const uint32_t ks2 = k0 ^ k1 ^ 0x1BD11BDAu;
    uint32_t v0 = x0 + k0;
    uint32_t v1 = x1 + k1;
    const int RA[4] = {13, 15, 26, 6};
    const int RB[4] = {17, 29, 16, 24};
#pragma unroll
    for (int g = 0; g < 5; ++g) {
#pragma unroll
        for (int i = 0; i < 4; ++i) {
            int r = (g & 1) ? RB[i] : RA[i];
            v0 += v1; v1 = rotl32(v1, r); v1 ^= v0;
        }
        switch (g) {
            case 0: v0 += k1;  v1 += ks2 + 1u; break;
            case 1: v0 += ks2; v1 += k0  + 2u; break;
            case 2: v0 += k0;  v1 += k1  + 3u; break;
            case 3: v0 += k1;  v1 += ks2 + 4u; break;
            case 4: v0 += ks2; v1 += k0  + 5u; break;
        }
    }
    o0 = v0; o1 = v1;
}

// bits element i of a length-L (even) counter draw, JAX half-split layout
__device__ __forceinline__ uint32_t jax_bits(uint32_t k0, uint32_t k1,
                                             uint32_t i, uint32_t L) {
    uint32_t H = L >> 1, o0, o1;
    if (i < H) { threefry2x32(k0, k1, i, i + H, o0, o1); return o0; }
    threefry2x32(k0, k1, i - H, i, o0, o1); return o1;
}

// key(42) -> split(B) -> split(2): per-batch (kc, kn)
__device__ __forceinline__ void batch_keys(int b, uint32_t& kc0, uint32_t& kc1,
                                           uint32_t& kn0, uint32_t& kn1) {
    uint32_t kb0 = jax_bits(0u, 42u, (uint32_t)(2 * b),     16u);
    uint32_t kb1 = jax_bits(0u, 42u, (uint32_t)(2 * b + 1), 16u);
    kc0 = jax_bits(kb0, kb1, 0u, 4u);
    kc1 = jax_bits(kb0, kb1, 1u, 4u);
    kn0 = jax_bits(kb0, kb1, 2u, 4u);
    kn1 = jax_bits(kb0, kb1, 3u, 4u);
}

// ---------------- voxel hash ----------------
__device__ __forceinline__ uint32_t vox_key(int vx, int vy, int vz) {
    return (uint32_t)(((vx + 512) * 1024 + (vy + 512)) * 1024 + (vz + 512));
}

__device__ __forceinline__ int lower_bound_u32(const uint32_t* a, int n, uint32_t v) {
    int lo = 0, hi = n;
    while (lo < hi) { int m = (lo + hi) >> 1; if (a[m] < v) lo = m + 1; else hi = m; }
    return lo;
}
__device__ __forceinline__ int upper_bound_u32(const uint32_t* a, int n, uint32_t v) {
    int lo = 0, hi = n;
    while (lo < hi) { int m = (lo + hi) >> 1; if (a[m] <= v) lo = m + 1; else hi = m; }
    return lo;
}

// ---------------- kernels ----------------
__global__ void k_zero(float* p, size_t n) {
    size_t t = (size_t)blockIdx.x * 256 + threadIdx.x;
    if (t < n) p[t] = 0.0f;
}

__global__ void k_keys(const float* __restrict__ pos,
                       uint32_t* __restrict__ keys, uint32_t* __restrict__ idx) {
    int tid = blockIdx.x * 256 + threadIdx.x;
    if (tid >= BB * NP) return;
    int i = tid % NP;
    const float* p = pos + (size_t)tid * 3;
    int vx = (int)floorf(p[0] * 4.0f);
    int vy = (int)floorf(p[1] * 4.0f);
    int vz = (int)floorf(p[2] * 4.0f);
    keys[tid] = vox_key(vx, vy, vz);
    idx[tid]  = (uint32_t)i;
}

// radix pass 1/3: per-tile digit histogram, digit-major layout hist[b][digit][tile]
__global__ void k_hist(const uint32_t* __restrict__ keys,
                       uint32_t* __restrict__ hist, int shift) {
    __shared__ uint32_t cnt[256];
    int b = blockIdx.y, tile = blockIdx.x, t = threadIdx.x;
    cnt[t] = 0;
    __syncthreads();
    int i = tile * 256 + t;
    if (i < NP) atomicAdd(&cnt[(keys[(size_t)b * NP + i] >> shift) & 255u], 1u);
    __syncthreads();
    hist[((size_t)b * 256 + t) * NT + tile] = cnt[t];
}

// radix pass 2/3: exclusive scan of 256*NT entries per batch (chunked Hillis-Steele)
__global__ void k_scan(uint32_t* __restrict__ hist) {
    __shared__ uint32_t tmp[256];
    __shared__ uint32_t carry;
    int b = blockIdx.x, t = threadIdx.x;
    if (t == 0) carry = 0;
    __syncthreads();
    uint32_t* h = hist + (size_t)b * 256 * NT;
    for (int c = 0; c < 256 * NT; c += 256) {
        uint32_t v = h[c + t];
        tmp[t] = v;
        __syncthreads();
        for (int off = 1; off < 256; off <<= 1) {
            uint32_t x = (t >= off) ? tmp[t - off] : 0u;
            __syncthreads();
            tmp[t] += x;
            __syncthreads();
        }
        uint32_t incl = tmp[t];
        uint32_t excl = incl - v + carry;     // carry read before update
        __syncthreads();
        if (t == 255) carry += incl;
        h[c + t] = excl;
        __syncthreads();
    }
}

// radix pass 3/3: stable scatter (serial in-tile rank => deterministic)
__global__ void k_scatter(const uint32_t* __restrict__ kin, const uint32_t* __restrict__ iin,
                          uint32_t* __restrict__ kout, uint32_t* __restrict__ iout,
                          const uint32_t* __restrict__ hist, int shift) {
    __shared__ int dls[256];
    int b = blockIdx.y, tile = blockIdx.x, t = threadIdx.x;
    int i = tile * 256 + t;
    uint32_t key = 0, idx = 0;
    int digit = -1;
    if (i < NP) {
        key = kin[(size_t)b * NP + i];
        idx = iin[(size_t)b * NP + i];
        digit = (int)((key >> shift) & 255u);
    }
    dls[t] = digit;
    __syncthreads();
    if (i < NP) {
        int rank = 0;
        for (int j = 0; j < t; ++j) rank += (dls[j] == digit);
        uint32_t dest = hist[((size_t)b * 256 + digit) * NT + tile] + (uint32_t)rank;
        kout[(size_t)b * NP + dest] = key;
        iout[(size_t)b * NP + dest] = idx;
    }
}

__global__ void k_centers(uint32_t* __restrict__ ctr,
                          float* __restrict__ o_ctr, float* __restrict__ o_cflag) {
    int tid = blockIdx.x * 256 + threadIdx.x;
    if (tid >= BB * NSAMP) return;
    int b = tid / NSAMP, i = tid % NSAMP;
    uint32_t kc0, kc1, kn0, kn1;
    batch_keys(b, kc0, kc1, kn0, kn1);
    uint32_t bits = jax_bits(kc0, kc1, (uint32_t)i, (uint32_t)NSAMP);
    uint32_t c = bits % (uint32_t)NP;
    ctr[tid] = c;
    o_ctr[tid] = (float)c;
    o_cflag[(size_t)b * NP + c] = 1.0f;
}

// one wave32 per (b, s): 27-lane bin search, 32-lane neighbor draw
__global__ __launch_bounds__(256) void k_query(
        const float* __restrict__ pos, const uint32_t* __restrict__ skeys,
        const uint32_t* __restrict__ order, const uint32_t* __restrict__ ctr,
        uint32_t* __restrict__ srcIdx, float* __restrict__ o_src,
        float* __restrict__ o_sflag) {
    __shared__ int loA[8][27];
    __shared__ int cumA[8][28];
    int w = threadIdx.x >> 5, lane = threadIdx.x & 31;
    int gw = blockIdx.x * 8 + w;
    int b = gw / NSAMP, s = gw % NSAMP;
    uint32_t c = ctr[(size_t)b * NSAMP + s];
    const float* pc = pos + ((size_t)b * NP + c) * 3;
    int vx = (int)floorf(pc[0] * 4.0f);
    int vy = (int)floorf(pc[1] * 4.0f);
    int vz = (int)floorf(pc[2] * 4.0f);
    const uint32_t* a = skeys + (size_t)b * NP;
    if (lane < 27) {
        int dx = lane / 9 - 1, dy = (lane / 3) % 3 - 1, dz = lane % 3 - 1;
        uint32_t nk = vox_key(vx + dx, vy + dy, vz + dz);
        int lo = lower_bound_u32(a, NP, nk);
        int hi = upper_bound_u32(a, NP, nk);
        loA[w][lane] = lo;
        cumA[w][lane + 1] = hi - lo;
    }
    __syncthreads();
    if (lane == 0) {
        cumA[w][0] = 0;
        for (int k = 1; k <= 27; ++k) cumA[w][k] += cumA[w][k - 1];
    }
    __syncthreads();
    int total = cumA[w][27];
    if (total < 1) total = 1;
    uint32_t kc0, kc1, kn0, kn1;
    batch_keys(b, kc0, kc1, kn0, kn1);
    uint32_t f = (uint32_t)(s * NNEI + lane);
    uint32_t rb = jax_bits(kn0, kn1, f, (uint32_t)(NSAMP * NNEI)) & 0x3FFFFFFFu;
    uint32_t r = rb % (uint32_t)total;
    int slot = 0;
    for (int k = 1; k <= 27; ++k) slot += (r >= (uint32_t)cumA[w][k]);
    int prev = cumA[w][slot];
    int start = loA[w][slot];
    uint32_t srci = order[(size_t)b * NP + start + ((int)r - prev)];
    size_t oidx = ((size_t)b * NSAMP + s) * NNEI + lane;
    srcIdx[oidx] = srci;
    o_src[oidx] = (float)srci;
    o_sflag[(size_t)b * NP + srci] = 1.0f;
}

// ---------------- host launcher ----------------
extern "C" void kernel_launch(void* const* d_in, const int* in_sizes, int n_in,
                              void* d_out, int out_size, void* d_ws, size_t ws_size,
                              hipStream_t stream) {
    const float* pos  = (const float*)d_in[0];
    const float* feat = (const float*)d_in[1];

    float* out    = (float*)d_out;
    float* o_src  = out    + (size_t)BB * NSAMP * NNEI * 67;
    float* o_ctr  = o_src  + (size_t)BB * NSAMP * NNEI;
    float* o_cfl  = o_ctr  + (size_t)BB * NSAMP;
    float* o_sfl  = o_cfl  + (size_t)BB * NP;   // contiguous with o_cfl

    uint32_t* keysA = (uint32_t*)d_ws;
    uint32_t* keysB = keysA + (size_t)BB * NP;
    uint32_t* idxA  = keysB + (size_t)BB * NP;
    uint32_t* idxB  = idxA  + (size_t)BB * NP;
    uint32_t* hist  = idxB  + (size_t)BB * NP;
    uint32_t* ctr   = hist  + (size_t)BB * 256 * NT;
    uint32_t* srcI  = ctr   + (size_t)BB * NSAMP;

    k_zero<<<(2 * BB * NP + 255) / 256, 256, 0, stream>>>(o_cfl, (size_t)2 * BB * NP);
    k_keys<<<(BB * NP + 255) / 256, 256, 0, stream>>>(pos, keysA, idxA);

    uint32_t *ka = keysA, *kb = keysB, *ia = idxA, *ib = idxB;
    for (int pass = 0; pass < 4; ++pass) {
        int shift = pass * 8;
        k_hist<<<dim3(NT, BB), 256, 0, stream>>>(ka, hist, shift);
        k_scan<<<BB, 256, 0, stream>>>(hist);
        k_scatter<<<dim3(NT, BB), 256, 0, stream>>>(ka, ia, kb, ib, hist, shift);
        uint32_t* t;
        t = ka; ka = kb; kb = t;
        t = ia; ia = ib; ib = t;
    }
    // after 4 passes: sorted keys in ka (== keysA), argsort order in ia (== idxA)

    k_centers<<<(BB * NSAMP + 255) / 256, 256, 0, stream>>>(ctr, o_ctr, o_cfl);
    k_query<<<BB * NSAMP / 8, 256, 0, stream>>>(pos, ka, ia, ctr, srcI, o_src, o_sfl);
    k_gather<<<BB * NSAMP / 8, 256, 0, stream>>>(pos, feat, ctr, srcI, out);
}